// HeteroGNN_30580167147609
// MI455X (gfx1250) — compile-verified
//
#include <hip/hip_runtime.h>

#define HD   64     // hidden / head dim
#define NH   4      // heads
#define FDIM 256    // NH*HD

typedef __attribute__((ext_vector_type(2))) float    v2f;
typedef __attribute__((ext_vector_type(8))) float    v8f;
typedef __attribute__((ext_vector_type(4))) unsigned u32x4;
typedef __attribute__((ext_vector_type(8))) int      i32x8;
typedef __attribute__((ext_vector_type(4))) int      i32x4;

// ---------------------------------------------------------------------------
// WMMA f32 GEMM:  C[M,N] (flags&2 ? += : =) A[M,K] @ W[K,N] (+bias) ; flags&1 -> relu
// Block = 4 waves = 4 consecutive 16-row tiles sharing one 64-col strip.
// The K x 64 panel of W is staged in LDS once per block via the Tensor Data
// Mover (TDM 2D descriptor, s_wait_tensorcnt), then each wave runs
// V_WMMA_F32_16X16X4_F32 with A from global (b64) and B from LDS.
// Requires K <= 256, K even, N % 64 == 0, M % 16 == 0 (true for all call sites).
// ---------------------------------------------------------------------------
__global__ __launch_bounds__(128) void gemm_f32_wmma(
    const float* __restrict__ A, const float* __restrict__ W,
    const float* __restrict__ bias, float* __restrict__ C,
    int M, int K, int N, int flags) {
  __shared__ float sW[256 * 64];               // K<=256 column panel, row stride 64
  const int tid  = threadIdx.x;
  const int lane = tid & 31;
  const int wave = tid >> 5;
  const int col0 = blockIdx.y << 6;

#if __has_builtin(__builtin_amdgcn_tensor_load_to_lds)
  if (wave == 0) {                              // one TDM issue per block
    const unsigned lds_off = (unsigned)(size_t)(void*)sW;
    const unsigned long long ga = (unsigned long long)(const void*)(W + col0);
    u32x4 g0;
    g0[0] = 1u;                                               // count=1, user D#
    g0[1] = lds_off;                                          // lds_addr (bytes)
    g0[2] = (unsigned)ga;                                     // global_addr[31:0]
    g0[3] = (unsigned)((ga >> 32) & 0x01FFFFFFull)            // global_addr[56:32]
            | 0x80000000u;                                    // type=2 ("image")
    i32x8 g1;
    g1[0] = (int)(2u << 16);                                  // data_size=4B
    g1[1] = (int)(((unsigned)N & 0xFFFFu) << 16);             // tensor_dim0[15:0]
    g1[2] = (int)((((unsigned)N >> 16) & 0xFFFFu)             // tensor_dim0[31:16]
                  | (((unsigned)K & 0xFFFFu) << 16));         // tensor_dim1[15:0]
    g1[3] = (int)((((unsigned)K >> 16) & 0xFFFFu)             // tensor_dim1[31:16]
                  | (64u << 16));                             // tile_dim0 = 64
    g1[4] = (int)((unsigned)K & 0xFFFFu);                     // tile_dim1=K, tile_dim2=0
    g1[5] = (int)N;                                           // tensor_dim0_stride[31:0]
    g1[6] = 0; g1[7] = 0;
    i32x4 gz = {0, 0, 0, 0};
#if __clang_major__ >= 23
    i32x8 gz8 = {0, 0, 0, 0, 0, 0, 0, 0};
    __builtin_amdgcn_tensor_load_to_lds(g0, g1, gz, gz, gz8, 0);
#else
    __builtin_amdgcn_tensor_load_to_lds(g0, g1, gz, gz, 0);
#endif
    __builtin_amdgcn_s_wait_tensorcnt(0);       // DMA landed before we barrier
  }
  __syncthreads();
#else
  for (int idx = tid; idx < K * 16; idx += 128) {
    const int k = idx >> 4, c4 = idx & 15;
    ((float4*)sW)[idx] = *(const float4*)(W + (size_t)k * N + col0 + c4 * 4);
  }
  __syncthreads();
#endif

  const int row0 = (blockIdx.x * 4 + wave) << 4;
  if (row0 >= M) return;                        // wave-uniform; after the barrier
  const int m = lane & 15;
  const int g = lane >> 4;
  v8f acc0 = {}, acc1 = {}, acc2 = {}, acc3 = {};
  const float* Arow = A + (size_t)(row0 + m) * K;

  const int Kmain = K & ~3;
  for (int k0 = 0; k0 < Kmain; k0 += 4) {       // unguarded hot loop
    const int ka = k0 + 2 * g;                  // even -> 8B-aligned (K even)
    const v2f a = *(const v2f*)(Arow + ka);     // global_load_b64
    const float* s0 = sW + ka * 64 + m;         // k row stride 64 in LDS
    v2f b0, b1, b2, b3;
    b0.x = s0[0];   b0.y = s0[64 + 0];
    b1.x = s0[16];  b1.y = s0[64 + 16];
    b2.x = s0[32];  b2.y = s0[64 + 32];
    b3.x = s0[48];  b3.y = s0[64 + 48];
    acc0 = __builtin_amdgcn_wmma_f32_16x16x4_f32(false, a, false, b0, (short)0, acc0, false, false);
    acc1 = __builtin_amdgcn_wmma_f32_16x16x4_f32(false, a, false, b1, (short)0, acc1, false, false);
    acc2 = __builtin_amdgcn_wmma_f32_16x16x4_f32(false, a, false, b2, (short)0, acc2, false, false);
    acc3 = __builtin_amdgcn_wmma_f32_16x16x4_f32(false, a, false, b3, (short)0, acc3, false, false);
  }
  if (Kmain < K) {                              // single guarded tail step (K=166 only)
    const int ka = Kmain + 2 * g;
    const bool ok0 = (ka < K), ok1 = (ka + 1 < K);
    v2f a;
    a.x = ok0 ? Arow[ka] : 0.0f;
    a.y = ok1 ? Arow[ka + 1] : 0.0f;
    const float* s0 = sW + ka * 64 + m;
    v2f b0, b1, b2, b3;
    b0.x = ok0 ? s0[0]  : 0.0f;  b0.y = ok1 ? s0[64 + 0]  : 0.0f;
    b1.x = ok0 ? s0[16] : 0.0f;  b1.y = ok1 ? s0[64 + 16] : 0.0f;
    b2.x = ok0 ? s0[32] : 0.0f;  b2.y = ok1 ? s0[64 + 32] : 0.0f;
    b3.x = ok0 ? s0[48] : 0.0f;  b3.y = ok1 ? s0[64 + 48] : 0.0f;
    acc0 = __builtin_amdgcn_wmma_f32_16x16x4_f32(false, a, false, b0, (short)0, acc0, false, false);
    acc1 = __builtin_amdgcn_wmma_f32_16x16x4_f32(false, a, false, b1, (short)0, acc1, false, false);
    acc2 = __builtin_amdgcn_wmma_f32_16x16x4_f32(false, a, false, b2, (short)0, acc2, false, false);
    acc3 = __builtin_amdgcn_wmma_f32_16x16x4_f32(false, a, false, b3, (short)0, acc3, false, false);
  }

  const v8f acc[4] = {acc0, acc1, acc2, acc3};
#pragma unroll
  for (int t = 0; t < 4; ++t) {
    const int col = col0 + t * 16 + m;
#pragma unroll
    for (int v = 0; v < 8; ++v) {
      const int r = row0 + v + 8 * g;
      const size_t idx = (size_t)r * N + col;
      float val = acc[t][v];
      if (bias) val += bias[col];
      if (flags & 2) val += C[idx];
      if (flags & 1) val = fmaxf(val, 0.0f);
      C[idx] = val;
    }
  }
}

// ---------------------------------------------------------------------------
// elementwise helpers
// ---------------------------------------------------------------------------
__global__ void fill_k(float* p, float v, size_t n) {
  size_t i = (size_t)blockIdx.x * blockDim.x + threadIdx.x;
  if (i < n) p[i] = v;
}
__global__ void relu_k(float* p, size_t n) {
  size_t i = (size_t)blockIdx.x * blockDim.x + threadIdx.x;
  if (i < n) p[i] = fmaxf(p[i], 0.0f);
}
// out = relu(agg + 0.5*(bA[col]+bB[col]))  (col = i % 256)
__global__ void gat_fin(float* agg, const float* __restrict__ bA,
                        const float* __restrict__ bB, size_t total) {
  size_t i = (size_t)blockIdx.x * blockDim.x + threadIdx.x;
  if (i >= total) return;
  int col = (int)(i & (FDIM - 1));
  agg[i] = fmaxf(agg[i] + 0.5f * (bA[col] + bB[col]), 0.0f);
}

// ---------------------------------------------------------------------------
// el[n,h] = sum_d h[n, h*64+d] * av[h*64+d]   (one wave per (n,h) task)
// ---------------------------------------------------------------------------
__global__ void row_head_dot(const float* __restrict__ h, const float* __restrict__ av,
                             float* __restrict__ out, int Nn) {
  const int lane = threadIdx.x & 31;
  const int wave = threadIdx.x >> 5;
  const int task = blockIdx.x * 4 + wave;
  if (task >= Nn * NH) return;
  const int n = task >> 2, hd = task & 3;
  const float* hr = h + (size_t)n * FDIM + hd * HD;
  const float* ar = av + hd * HD;
  float v = hr[lane] * ar[lane] + hr[lane + 32] * ar[lane + 32];
#pragma unroll
  for (int o = 16; o; o >>= 1) v += __shfl_xor(v, o, 32);
  if (lane == 0) out[task] = v;
}

// ---------------------------------------------------------------------------
// edge kernels (GAT softmax)
// ---------------------------------------------------------------------------
__device__ __forceinline__ unsigned f2key(float x) {
  unsigned k = __float_as_uint(x);
  return (k & 0x80000000u) ? ~k : (k | 0x80000000u);
}
__global__ void edge_score(const int* __restrict__ src, const int* __restrict__ dst,
                           const float* __restrict__ el, const float* __restrict__ er,
                           float* __restrict__ e, unsigned* __restrict__ mkey, int E) {
  int i = blockIdx.x * blockDim.x + threadIdx.x;
  if (i >= E) return;
  int s = src[i], d = dst[i];
  float4 a = ((const float4*)el)[s];
  float4 b = ((const float4*)er)[d];
  float v[4] = {a.x + b.x, a.y + b.y, a.z + b.z, a.w + b.w};
  float4 o;
#pragma unroll
  for (int h = 0; h < 4; ++h) {
    float x = v[h];
    x = (x > 0.0f) ? x : 0.2f * x;                 // leaky_relu(0.2)
    ((float*)&o)[h] = x;
    atomicMax(&mkey[(size_t)d * 4 + h], f2key(x)); // ordered-uint segment max
  }
  ((float4*)e)[i] = o;
}
__global__ void unmap_k(unsigned* m, int n) {
  int i = blockIdx.x * blockDim.x + threadIdx.x;
  if (i >= n) return;
  unsigned k = m[i];
  m[i] = (k & 0x80000000u) ? (k & 0x7FFFFFFFu) : ~k;  // back to float bits
}
__global__ void edge_exp(const int* __restrict__ dst, float* __restrict__ e,
                         const float* __restrict__ m, float* __restrict__ ssum, int E) {
  int i = blockIdx.x * blockDim.x + threadIdx.x;
  if (i >= E) return;
  int d = dst[i];
  float4 e4 = ((const float4*)e)[i];
  float4 m4 = ((const float4*)m)[d];
  float4 o;
  o.x = __expf(e4.x - m4.x); o.y = __expf(e4.y - m4.y);
  o.z = __expf(e4.z - m4.z); o.w = __expf(e4.w - m4.w);
  ((float4*)e)[i] = o;
  atomicAdd(&ssum[(size_t)d * 4 + 0], o.x);
  atomicAdd(&ssum[(size_t)d * 4 + 1], o.y);
  atomicAdd(&ssum[(size_t)d * 4 + 2], o.z);
  atomicAdd(&ssum[(size_t)d * 4 + 3], o.w);
}
// one wave per edge: agg[dst, :] += coef * a_h * h_src[src, :]
__global__ void edge_agg(const int* __restrict__ src, const int* __restrict__ dst,
                         const float* __restrict__ ex, const float* __restrict__ ssum,
                         const float* __restrict__ hsrc, float* __restrict__ agg,
                         float coef, int E) {
  const int lane = threadIdx.x & 31;
  const int ei = (int)(((size_t)blockIdx.x * blockDim.x + threadIdx.x) >> 5);
  if (ei >= E) return;
  int s = src[ei], d = dst[ei];
  float4 x4 = ((const float4*)ex)[ei];
  float4 s4 = ((const float4*)ssum)[d];
  float aa[4] = {coef * x4.x / s4.x, coef * x4.y / s4.y,
                 coef * x4.z / s4.z, coef * x4.w / s4.w};
  const float* h = hsrc + (size_t)s * FDIM;
  float* o = agg + (size_t)d * FDIM;
#pragma unroll
  for (int j = 0; j < 8; ++j) {
    int col = lane + 32 * j;
    atomicAdd(&o[col], aa[j >> 1] * h[col]);   // head = col/64 = j/2 (compile-time)
  }
}

// ---------------------------------------------------------------------------
// SAGE edge kernels
// ---------------------------------------------------------------------------
__global__ void sage_edge(const int* __restrict__ src, const int* __restrict__ dst,
                          const float* __restrict__ x, float* __restrict__ hn,
                          float* __restrict__ deg, int E) {
  const int lane = threadIdx.x & 31;
  const int ei = (int)(((size_t)blockIdx.x * blockDim.x + threadIdx.x) >> 5);
  if (ei >= E) return;
  int s = src[ei], d = dst[ei];
  if (lane == 0) atomicAdd(&deg[d], 1.0f);
  const float* xr = x + (size_t)s * FDIM;
  float* o = hn + (size_t)d * FDIM;
#pragma unroll
  for (int j = 0; j < 8; ++j) {
    int col = lane + 32 * j;
    atomicAdd(&o[col], xr[col]);
  }
}
__global__ void hn_div(float* hn, const float* __restrict__ deg, size_t total) {
  size_t i = (size_t)blockIdx.x * blockDim.x + threadIdx.x;
  if (i >= total) return;
  hn[i] /= fmaxf(deg[i >> 8], 1.0f);
}

// final 256->2 SAGE layer (writes d_out directly; no relu on last layer)
__global__ void sage2_k(const float* __restrict__ x, const float* __restrict__ Ws,
                        const float* __restrict__ hn, const float* __restrict__ Wn,
                        const float* __restrict__ b, float* __restrict__ out,
                        int Nn, int acc) {
  int n = blockIdx.x * blockDim.x + threadIdx.x;
  if (n >= Nn) return;
  float o0 = b[0], o1 = b[1];
  if (acc) { o0 += out[(size_t)n * 2]; o1 += out[(size_t)n * 2 + 1]; }
  const float* xr = x + (size_t)n * FDIM;
  const float* hr = hn + (size_t)n * FDIM;
  for (int k = 0; k < FDIM; ++k) {
    float xv = xr[k], hv = hr[k];
    o0 += xv * Ws[k * 2 + 0] + hv * Wn[k * 2 + 0];
    o1 += xv * Ws[k * 2 + 1] + hv * Wn[k * 2 + 1];
  }
  out[(size_t)n * 2] = o0;
  out[(size_t)n * 2 + 1] = o1;
}

// ---------------------------------------------------------------------------
static inline int cdiv(long long a, long long b) { return (int)((a + b - 1) / b); }

extern "C" void kernel_launch(void* const* d_in, const int* in_sizes, int n_in,
                              void* d_out, int out_size, void* d_ws, size_t ws_size,
                              hipStream_t stream) {
  const float* tx_feat = (const float*)d_in[0];
  const float* w_feat  = (const float*)d_in[1];
  const int*   edges   = (const int*)d_in[2];
  const float* tx_W = (const float*)d_in[3];
  const float* tx_b = (const float*)d_in[4];
  const float* w_W  = (const float*)d_in[5];
  const float* w_b  = (const float*)d_in[6];
  const float* g1W  = (const float*)d_in[7];
  const float* g1al = (const float*)d_in[8];
  const float* g1ar = (const float*)d_in[9];
  const float* g1b  = (const float*)d_in[10];
  const float* s1Ws = (const float*)d_in[11];
  const float* s1Wn = (const float*)d_in[12];
  const float* s1b  = (const float*)d_in[13];
  const float* g2W  = (const float*)d_in[14];
  const float* g2al = (const float*)d_in[15];
  const float* g2ar = (const float*)d_in[16];
  const float* g2b  = (const float*)d_in[17];
  const float* s2Ws = (const float*)d_in[18];
  const float* s2Wn = (const float*)d_in[19];
  const float* s2b  = (const float*)d_in[20];

  const int NTX = in_sizes[0] / 166;
  const int NW  = in_sizes[1] / 56;
  const int E   = in_sizes[2] / 8;     // (4,2,E)
  const int NS  = NTX > NW ? NTX : NW;

  // scratch arenas (~241 MB total)
  float* ws = (float*)d_ws;
  size_t off = 0;
  auto alloc = [&](size_t n) { float* p = ws + off; off += (n + 63) & ~(size_t)63; return p; };
  float* P0  = alloc((size_t)NTX * HD);    // t_in / sage1_tx out
  float* P1  = alloc((size_t)NW * HD);     // w_in / sage1_w out
  float* P2  = alloc((size_t)NS * FDIM);   // h_src / hn scratch
  float* P3  = alloc((size_t)NS * FDIM);   // h_dst scratch
  float* P4  = alloc((size_t)NTX * FDIM);  // GAT agg tx
  float* P5  = alloc((size_t)NW * FDIM);   // GAT agg w
  float* elb = alloc((size_t)NS * 4);
  float* erb = alloc((size_t)NS * 4);
  float* mb  = alloc((size_t)NS * 4);
  float* sb  = alloc((size_t)NS * 4);
  float* deg = alloc((size_t)NS);
  float* eb  = alloc((size_t)E * 4);
  (void)ws_size; (void)n_in; (void)out_size;

  auto fill = [&](float* p, float v, size_t n) {
    fill_k<<<cdiv((long long)n, 256), 256, 0, stream>>>(p, v, n);
  };
  auto gemm = [&](const float* A, const float* Wm, const float* b, float* C,
                  int M, int K, int N, int flags) {
    dim3 grid(cdiv(cdiv(M, 16), 4), N / 64);
    gemm_f32_wmma<<<grid, 128, 0, stream>>>(A, Wm, b, C, M, K, N, flags);
  };

  auto gat_layer = [&](const float* tin, const float* win, const float* Wt,
                       const float* al, const float* ar, const float* bb) {
    fill(P4, 0.0f, (size_t)NTX * FDIM);
    fill(P5, 0.0f, (size_t)NW * FDIM);
    for (int r = 0; r < 4; ++r) {
      const bool toTx = ((r & 1) == 0);          // r=0,2: w->tx ; r=1,3: tx->w
      const float* Sx = toTx ? win : tin;
      const float* Dx = toTx ? tin : win;
      const int Ns = toTx ? NW : NTX;
      const int Nd = toTx ? NTX : NW;
      float* aggD = toTx ? P4 : P5;
      const int* srcp = edges + (size_t)(2 * r) * E;
      const int* dstp = edges + (size_t)(2 * r + 1) * E;
      const float* Wr = Wt + (size_t)r * HD * FDIM;
      gemm(Sx, Wr, nullptr, P2, Ns, HD, FDIM, 0);            // h_s
      gemm(Dx, Wr, nullptr, P3, Nd, HD, FDIM, 0);            // h_d
      row_head_dot<<<cdiv(Ns * NH, 4), 128, 0, stream>>>(P2, al + (size_t)r * FDIM, elb, Ns);
      row_head_dot<<<cdiv(Nd * NH, 4), 128, 0, stream>>>(P3, ar + (size_t)r * FDIM, erb, Nd);
      fill(mb, 0.0f, (size_t)Nd * 4);                        // key 0 == -inf
      fill(sb, 0.0f, (size_t)Nd * 4);
      edge_score<<<cdiv(E, 256), 256, 0, stream>>>(srcp, dstp, elb, erb, eb, (unsigned*)mb, E);
      unmap_k<<<cdiv(Nd * 4, 256), 256, 0, stream>>>((unsigned*)mb, Nd * 4);
      edge_exp<<<cdiv(E, 256), 256, 0, stream>>>(dstp, eb, mb, sb, E);
      edge_agg<<<cdiv((long long)E * 32, 256), 256, 0, stream>>>(srcp, dstp, eb, sb, P2, aggD, 0.5f, E);
    }
    gat_fin<<<cdiv((long long)NTX * FDIM, 256), 256, 0, stream>>>(P4, bb, bb + 2 * FDIM, (size_t)NTX * FDIM);
    gat_fin<<<cdiv((long long)NW * FDIM, 256), 256, 0, stream>>>(P5, bb + FDIM, bb + 3 * FDIM, (size_t)NW * FDIM);
  };

  auto sage_layer = [&](const float* Wsm, const float* Wnm, const float* bb) {
    for (int r = 0; r < 4; ++r) {
      const bool toTx = ((r & 1) == 0);
      const float* Sx = toTx ? P5 : P4;
      const float* Dx = toTx ? P4 : P5;
      const int Nd = toTx ? NTX : NW;
      float* outp = toTx ? P0 : P1;
      const int* srcp = edges + (size_t)(2 * r) * E;
      const int* dstp = edges + (size_t)(2 * r + 1) * E;
      fill(deg, 0.0f, (size_t)Nd);
      fill(P2, 0.0f, (size_t)Nd * FDIM);
      sage_edge<<<cdiv((long long)E * 32, 256), 256, 0, stream>>>(srcp, dstp, Sx, P2, deg, E);
      hn_div<<<cdiv((long long)Nd * FDIM, 256), 256, 0, stream>>>(P2, deg, (size_t)Nd * FDIM);
      const int base = (r < 2) ? 0 : 2;                      // first write vs accumulate
      gemm(Dx, Wsm + (size_t)r * FDIM * HD, bb + (size_t)r * HD, outp, Nd, FDIM, HD, base);
      gemm(P2, Wnm + (size_t)r * FDIM * HD, nullptr, outp, Nd, FDIM, HD, 2);
    }
    relu_k<<<cdiv((long long)NTX * HD, 256), 256, 0, stream>>>(P0, (size_t)NTX * HD);
    relu_k<<<cdiv((long long)NW * HD, 256), 256, 0, stream>>>(P1, (size_t)NW * HD);
  };

  // ---- forward pass ----
  gemm(tx_feat, tx_W, tx_b, P0, NTX, 166, HD, 1);            // relu(tx_feat @ tx_W + b)
  gemm(w_feat,  w_W,  w_b,  P1, NW,  56,  HD, 1);
  gat_layer(P0, P1, g1W, g1al, g1ar, g1b);                   // -> P4 (tx,256), P5 (w,256), relu'd
  sage_layer(s1Ws, s1Wn, s1b);                               // -> P0 (tx,64), P1 (w,64), relu'd
  gat_layer(P0, P1, g2W, g2al, g2ar, g2b);                   // -> P4, P5
  // SAGE2 (256 -> 2), writes straight into d_out: [t (NTX,2) | w (NW,2)]
  float* outT = (float*)d_out;
  float* outW = outT + (size_t)NTX * 2;
  for (int r = 0; r < 4; ++r) {
    const bool toTx = ((r & 1) == 0);
    const float* Sx = toTx ? P5 : P4;
    const float* Dx = toTx ? P4 : P5;
    const int Nd = toTx ? NTX : NW;
    float* outp = toTx ? outT : outW;
    const int* srcp = edges + (size_t)(2 * r) * E;
    const int* dstp = edges + (size_t)(2 * r + 1) * E;
    fill(deg, 0.0f, (size_t)Nd);
    fill(P2, 0.0f, (size_t)Nd * FDIM);
    sage_edge<<<cdiv((long long)E * 32, 256), 256, 0, stream>>>(srcp, dstp, Sx, P2, deg, E);
    hn_div<<<cdiv((long long)Nd * FDIM, 256), 256, 0, stream>>>(P2, deg, (size_t)Nd * FDIM);
    sage2_k<<<cdiv(Nd, 256), 256, 0, stream>>>(Dx, s2Ws + (size_t)r * FDIM * 2, P2,
                                               s2Wn + (size_t)r * FDIM * 2, s2b + r * 2,
                                               outp, Nd, (r < 2) ? 0 : 1);
  }
}